// FourierModel_70196945486447
// MI455X (gfx1250) — compile-verified
//
#include <hip/hip_runtime.h>
#include <hip/hip_bf16.h>
#include <math.h>

typedef __attribute__((ext_vector_type(2))) float v2f;
typedef __attribute__((ext_vector_type(8))) float v8f;
typedef __attribute__((ext_vector_type(4))) unsigned int u32x4;
typedef __attribute__((ext_vector_type(4))) int i32x4;
typedef __attribute__((ext_vector_type(8))) int i32x8;

#define S_SRC 2048
#define WDIM 1024
#define OUTDIM 4096
#define ORDER 32

#if defined(__has_builtin)
#if __has_builtin(__builtin_amdgcn_tensor_load_to_lds) && __has_builtin(__builtin_amdgcn_s_wait_tensorcnt)
#define HAVE_TDM 1
#endif
#endif
#ifndef HAVE_TDM
#define HAVE_TDM 0
#endif

__device__ __forceinline__ float gelu_f(float x) {
    // tanh approximation (jax.nn.gelu default approximate=True)
    const float k0 = 0.7978845608028654f; // sqrt(2/pi)
    const float k1 = 0.044715f;
    float x3 = x * x * x;
    float t = tanhf(k0 * (x + k1 * x3));
    return 0.5f * x * (1.0f + t);
}

#if HAVE_TDM
// Flat LDS pointers encode the LDS byte offset in addr[31:0] (ISA 10.2 aperture
// mapping), so truncation yields the D# lds_addr field.
__device__ __forceinline__ unsigned lds_byte_off(const void* p) {
    return (unsigned)(unsigned long long)(uintptr_t)p;
}

__device__ __forceinline__ void tdm_raw(u32x4 g0, i32x8 g1) {
#if defined(__clang_major__) && (__clang_major__ >= 23)
    i32x4 z4 = {};
    i32x8 z8 = {};
    __builtin_amdgcn_tensor_load_to_lds(g0, g1, z4, z4, z8, 0);
#else
    i32x4 z4 = {};
    __builtin_amdgcn_tensor_load_to_lds(g0, g1, z4, z4, 0);
#endif
}

// Issue a TDM DMA of a (tile_h x 64) f32 tile (row stride = rowStride elems)
// into LDS with 1 DWORD of padding inserted after every 64 DWORDs
// (pad_interval=5 -> 64 DWORDs, pad_amount=0 -> 1 DWORD), matching the
// [rows][64+1] LDS layout used by the WMMA fragment reads.
__device__ __forceinline__ void tdm_issue_2d(const float* gptr, unsigned ldsoff,
                                             int tile_h, int rowStride) {
    unsigned long long ga = (unsigned long long)(uintptr_t)gptr;
    u32x4 g0;
    g0[0] = 0x1u;                                    // count=1, user mode
    g0[1] = ldsoff;                                  // lds_addr (bytes)
    g0[2] = (unsigned)(ga & 0xFFFFFFFFu);            // global_addr[31:0]
    g0[3] = (unsigned)((ga >> 32) & 0x01FFFFFFu)     // global_addr[56:32]
          | (2u << 30);                              // type = 2 (image/D#)
    const unsigned td = 1u << 30;                    // huge tensor dims: no clip
    i32x8 g1;
    g1[0] = (int)((2u << 16)                         // data_size = 4 bytes
                | (1u << 20)                         // pad_enable
                | (5u << 22));                       // pad_interval: 64 DWORDs
    g1[1] = (int)((td & 0xFFFFu) << 16);             // tensor_dim0 lo16
    g1[2] = (int)((td >> 16) | ((td & 0xFFFFu) << 16));   // td0 hi / td1 lo
    g1[3] = (int)((td >> 16) | (64u << 16));         // td1 hi | tile_dim0 = 64
    g1[4] = (int)(tile_h & 0xFFFF);                  // tile_dim1, tile_dim2 = 0
    g1[5] = rowStride;                               // tensor_dim0_stride[31:0]
    g1[6] = 0;                                       // stride hi, dim1_stride lo
    g1[7] = 0;
    tdm_raw(g0, g1);
}
#endif  // HAVE_TDM

// ---------------------------------------------------------------------------
// Layer 1: h1 = gelu(src @ W1.T + b1), K = 4 (scalar, trivial FLOPs)
// ---------------------------------------------------------------------------
__global__ __launch_bounds__(256) void layer1_kernel(
    const float* __restrict__ src, const float* __restrict__ W1,
    const float* __restrict__ b1, float* __restrict__ H) {
    int idx = blockIdx.x * 256 + threadIdx.x;           // s * WDIM + j
    int s = idx >> 10;
    int j = idx & (WDIM - 1);
    float a = b1[j];
    const float* sr = src + s * 4;
    const float* wr = W1 + j * 4;
    a += sr[0] * wr[0] + sr[1] * wr[1] + sr[2] * wr[2] + sr[3] * wr[3];
    H[idx] = gelu_f(a);
}

// ---------------------------------------------------------------------------
// WMMA f32 GEMM: C[s,j] = gelu(sum_k A[s,k] * Wm[j,k] + b[j])
// Block = 256 threads = 8 waves; block tile 128(M) x 16(N); K chunks of 64.
// Each wave computes one 16x16 C tile with V_WMMA_F32_16X16X4_F32.
// LDS staging via Tensor Data Mover (per-wave 16x64 A slices + shared B tile),
// double-buffered across K-chunks with S_WAIT_TENSORCNT.
// ---------------------------------------------------------------------------
#define TSM 128
#define TSN 16
#define KC  64

__global__ __launch_bounds__(256) void gemm_gelu_wmma(
    const float* __restrict__ A,    // S x K
    const float* __restrict__ Wm,   // Nout x K (we multiply by Wm^T)
    const float* __restrict__ bias, // Nout
    float* __restrict__ C,          // S x Nout
    int K, int Nout) {
#if HAVE_TDM
    __shared__ float Asub[2][TSM][KC + 1];
    __shared__ float Bsub[2][TSN][KC + 1];
#else
    __shared__ float Asub[1][TSM][KC + 1];
    __shared__ float Bsub[1][TSN][KC + 1];
#endif

    const int tid  = threadIdx.x;
    const int wave = tid >> 5;
    const int lane = tid & 31;
    const int rowBase = blockIdx.x * TSM;
    const int colBase = blockIdx.y * TSN;

    const int hi = lane >> 4;     // 0: K-pair {0,1}, 1: K-pair {2,3}
    const int rl = lane & 15;     // row (A) / col (B) within tile
    const int chunks = K / KC;

    v8f acc = {};

#if HAVE_TDM
    const float* aSlice = A + (size_t)(rowBase + wave * 16) * K;  // this wave's rows
    const float* bBase  = Wm + (size_t)colBase * K;

    // Prologue: DMA chunk 0 into buffer 0.
    tdm_issue_2d(aSlice, lds_byte_off(&Asub[0][wave * 16][0]), 16, K);
    if (wave == 0) tdm_issue_2d(bBase, lds_byte_off(&Bsub[0][0][0]), TSN, K);

    for (int c = 0; c < chunks; ++c) {
        const int cur = c & 1;
        if (c + 1 < chunks) {
            const int nxt = cur ^ 1;
            // Issue next chunk's DMA, then wait for the current chunk only
            // (per-wave TENSORcnt is in-order: <=inflight(next) => cur done).
            tdm_issue_2d(aSlice + (size_t)(c + 1) * KC,
                         lds_byte_off(&Asub[nxt][wave * 16][0]), 16, K);
            if (wave == 0) {
                tdm_issue_2d(bBase + (size_t)(c + 1) * KC,
                             lds_byte_off(&Bsub[nxt][0][0]), TSN, K);
                __builtin_amdgcn_s_wait_tensorcnt(2);
            } else {
                __builtin_amdgcn_s_wait_tensorcnt(1);
            }
        } else {
            __builtin_amdgcn_s_wait_tensorcnt(0);
        }
        __syncthreads();   // publish B (wave 0's DMA) to all waves

        const int waveRow = wave * 16 + rl;
#pragma unroll
        for (int kk = 0; kk < KC; kk += 4) {
            v2f af, bf;
            af.x = Asub[cur][waveRow][kk + 2 * hi + 0];
            af.y = Asub[cur][waveRow][kk + 2 * hi + 1];
            bf.x = Bsub[cur][rl][kk + 2 * hi + 0];
            bf.y = Bsub[cur][rl][kk + 2 * hi + 1];
            acc = __builtin_amdgcn_wmma_f32_16x16x4_f32(
                false, af, false, bf, (short)0, acc, false, false);
        }
        __syncthreads();   // everyone done with buf[cur] before it is re-filled
    }
#else
    for (int k0 = 0; k0 < K; k0 += KC) {
        for (int i = tid; i < TSM * KC; i += 256) {
            int rr = i >> 6, kk = i & (KC - 1);
            Asub[0][rr][kk] = A[(rowBase + rr) * K + k0 + kk];
        }
        for (int i = tid; i < TSN * KC; i += 256) {
            int jj = i >> 6, kk = i & (KC - 1);
            Bsub[0][jj][kk] = Wm[(colBase + jj) * K + k0 + kk];
        }
        __syncthreads();
        const int waveRow = wave * 16 + rl;
#pragma unroll
        for (int kk = 0; kk < KC; kk += 4) {
            v2f af, bf;
            af.x = Asub[0][waveRow][kk + 2 * hi + 0];
            af.y = Asub[0][waveRow][kk + 2 * hi + 1];
            bf.x = Bsub[0][rl][kk + 2 * hi + 0];
            bf.y = Bsub[0][rl][kk + 2 * hi + 1];
            acc = __builtin_amdgcn_wmma_f32_16x16x4_f32(
                false, af, false, bf, (short)0, acc, false, false);
        }
        __syncthreads();
    }
#endif

    // C/D 16x16 f32 layout: VGPR v -> M = v (lanes 0-15) / 8+v (lanes 16-31)
    const int col = colBase + rl;
    const float bv = bias[col];
#pragma unroll
    for (int v = 0; v < 8; ++v) {
        int row = rowBase + wave * 16 + hi * 8 + v;
        C[row * Nout + col] = gelu_f(acc[v] + bv);
    }
}

// ---------------------------------------------------------------------------
// hsum[j] = sum_s H[s, j]   (coalesced column reduce)
// ---------------------------------------------------------------------------
__global__ __launch_bounds__(256) void reduce_cols_kernel(
    const float* __restrict__ H, float* __restrict__ hsum) {
    int j = blockIdx.x * 256 + threadIdx.x;
    float a = 0.f;
    for (int s = 0; s < S_SRC; ++s) a += H[s * WDIM + j];
    hsum[j] = a;
}

// ---------------------------------------------------------------------------
// bsum = (sum_s src[s,:]) . Wb + S*bb
// ---------------------------------------------------------------------------
__global__ __launch_bounds__(256) void bsum_kernel(
    const float* __restrict__ src, const float* __restrict__ Wb,
    const float* __restrict__ bb, float* __restrict__ bsum) {
    __shared__ float sh[256][4];
    int tid = threadIdx.x;
    float p0 = 0.f, p1 = 0.f, p2 = 0.f, p3 = 0.f;
    for (int s = tid; s < S_SRC; s += 256) {
        p0 += src[s * 4 + 0];
        p1 += src[s * 4 + 1];
        p2 += src[s * 4 + 2];
        p3 += src[s * 4 + 3];
    }
    sh[tid][0] = p0; sh[tid][1] = p1; sh[tid][2] = p2; sh[tid][3] = p3;
    __syncthreads();
    if (tid == 0) {
        float t0 = 0.f, t1 = 0.f, t2 = 0.f, t3 = 0.f;
        for (int i = 0; i < 256; ++i) {
            t0 += sh[i][0]; t1 += sh[i][1]; t2 += sh[i][2]; t3 += sh[i][3];
        }
        bsum[0] = t0 * Wb[0] + t1 * Wb[1] + t2 * Wb[2] + t3 * Wb[3]
                + (float)S_SRC * bb[0];
    }
}

// ---------------------------------------------------------------------------
// wsum[o] = W4[o,:] . hsum + S*b4[o]   (layer-4 collapsed to a matvec)
// ---------------------------------------------------------------------------
__global__ __launch_bounds__(256) void wsum_kernel(
    const float* __restrict__ W4, const float* __restrict__ b4,
    const float* __restrict__ hsum, float* __restrict__ wsum) {
    int o = blockIdx.x * 256 + threadIdx.x;
    const float* wr = W4 + (size_t)o * WDIM;
    float a = 0.f;
    for (int k = 0; k < WDIM; ++k) a += wr[k] * hsum[k];
    wsum[o] = a + (float)S_SRC * b4[o];
}

// ---------------------------------------------------------------------------
// Assemble the 64x64 bilinear-form matrix M = [[W0, W2], [W3, W1]]
// rows: [cx(0..31); sx(0..31)], cols: [cy(0..31); sy(0..31)]
// ---------------------------------------------------------------------------
__global__ __launch_bounds__(256) void buildM_kernel(
    const float* __restrict__ wsum, float* __restrict__ M) {
    int idx = blockIdx.x * 256 + threadIdx.x;   // p*64 + q
    int p = idx >> 6, q = idx & 63;
    int i = p & 31, j = q & 31;
    int c = (p < 32) ? ((q < 32) ? 0 : 2) : ((q < 32) ? 3 : 1);
    M[idx] = wsum[c * (ORDER * ORDER) + i * ORDER + j];
}

// ---------------------------------------------------------------------------
// out[n] = bsum + u_x^T M u_y,  u = [cos(w_i t); sin(w_i t)]
// M cached in LDS (uniform broadcast reads). y-sincos cached in registers;
// x-sincos recomputed inside the i-loop to keep VGPR pressure bounded.
// ---------------------------------------------------------------------------
__global__ __launch_bounds__(256) void fourier_kernel(
    const float* __restrict__ r, const float* __restrict__ M,
    const float* __restrict__ bsum, float* __restrict__ out, int N) {
    __shared__ float Ml[64 * 64];
    for (int i = threadIdx.x; i < 64 * 64; i += 256) Ml[i] = M[i];
    __syncthreads();

    int n = blockIdx.x * 256 + threadIdx.x;
    if (n >= N) return;

    const float w0 = 0.6283185307179586f;  // 2*pi/10
    float x = r[2 * n + 0];
    float y = r[2 * n + 1];

    float cy[32], sy[32];
#pragma unroll
    for (int j = 0; j < 32; ++j) {
        sincosf(w0 * (float)(j + 1) * y, &sy[j], &cy[j]);
    }

    float acc = 0.f;
    for (int i = 0; i < 32; ++i) {
        float sxv, cxv;
        sincosf(w0 * (float)(i + 1) * x, &sxv, &cxv);
        const float* rowC = Ml + i * 64;         // cx row
        const float* rowS = Ml + (32 + i) * 64;  // sx row
        float d0 = 0.f, d1 = 0.f;
#pragma unroll
        for (int j = 0; j < 32; ++j) {
            d0 = fmaf(rowC[j],      cy[j], d0);
            d0 = fmaf(rowC[32 + j], sy[j], d0);
            d1 = fmaf(rowS[j],      cy[j], d1);
            d1 = fmaf(rowS[32 + j], sy[j], d1);
        }
        acc = fmaf(cxv, d0, fmaf(sxv, d1, acc));
    }
    out[n] = bsum[0] + acc;
}

// ---------------------------------------------------------------------------
extern "C" void kernel_launch(void* const* d_in, const int* in_sizes, int n_in,
                              void* d_out, int out_size, void* d_ws, size_t ws_size,
                              hipStream_t stream) {
    const float* src = (const float*)d_in[0];   // (2048, 4)
    const float* r   = (const float*)d_in[1];   // (N, 2)
    const float* W1  = (const float*)d_in[2];
    const float* b1  = (const float*)d_in[3];
    const float* W2  = (const float*)d_in[4];
    const float* b2  = (const float*)d_in[5];
    const float* W3  = (const float*)d_in[6];
    const float* b3  = (const float*)d_in[7];
    const float* W4  = (const float*)d_in[8];
    const float* b4  = (const float*)d_in[9];
    const float* Wb  = (const float*)d_in[10];
    const float* bb  = (const float*)d_in[11];
    float* out = (float*)d_out;
    const int N = in_sizes[1] / 2;

    float* ws = (float*)d_ws;
    float* hA   = ws;                                   // 2048*1024
    float* hB   = hA + (size_t)S_SRC * WDIM;            // 2048*1024
    float* hsum = hB + (size_t)S_SRC * WDIM;            // 1024
    float* wsum = hsum + WDIM;                          // 4096
    float* Mmat = wsum + OUTDIM;                        // 4096
    float* bs   = Mmat + 64 * 64;                       // 1

    // Layer 1 (K=4): src -> hA
    layer1_kernel<<<(S_SRC * WDIM) / 256, 256, 0, stream>>>(src, W1, b1, hA);

    // Layers 2, 3: WMMA f32 GEMMs (2048 x 1024 x 1024 each)
    dim3 g2(S_SRC / TSM, WDIM / TSN);
    gemm_gelu_wmma<<<g2, 256, 0, stream>>>(hA, W2, b2, hB, WDIM, WDIM);
    gemm_gelu_wmma<<<g2, 256, 0, stream>>>(hB, W3, b3, hA, WDIM, WDIM);

    // hsum = sum_s h3[s,:]
    reduce_cols_kernel<<<WDIM / 256, 256, 0, stream>>>(hA, hsum);

    // bsum
    bsum_kernel<<<1, 256, 0, stream>>>(src, Wb, bb, bs);

    // wsum = W4 @ hsum + S*b4  (layer 4 collapsed through the source-sum)
    wsum_kernel<<<OUTDIM / 256, 256, 0, stream>>>(W4, b4, hsum, wsum);

    // M assembly
    buildM_kernel<<<(64 * 64) / 256, 256, 0, stream>>>(wsum, Mmat);

    // Fourier evaluation
    fourier_kernel<<<(N + 255) / 256, 256, 0, stream>>>(r, Mmat, bs, out, N);
}